// BatchNormSPD_35261681500475
// MI455X (gfx1250) — compile-verified
//
#include <hip/hip_runtime.h>
#include <math.h>

// ---------------------------------------------------------------------------
// BatchNormSPD forward for MI455X (gfx1250).
// One workgroup (256 threads = 8 wave32) owns one 64x64 SPD matrix in LDS;
// every 64x64 matmul runs on V_WMMA_F32_16X16X4_F32. The single GEMM routine
// computes C = A * B^T with BOTH operand fragments read row-major (contiguous
// ds_load_b64, conflict-free thanks to the stride-68 pad). Every B operand in
// this pipeline is symmetric (polynomials of symmetric matrices) or genuinely
// wanted transposed (final T*W^T), so A*B^T serves every call site with zero
// register repacking. Each wave computes two 16x16 tiles in one M-row so one
// A-fragment feeds two WMMAs. Matrix sqrt/invsqrt/log/exp are eigen-free
// Newton-Schulz / scaling-and-squaring schemes built purely from these WMMA
// GEMMs. The batched matrix-log reduction uses 32-way partial accumulators to
// avoid a global_atomic_add_f32 hotspot.
// ---------------------------------------------------------------------------

#define TPB   256
#define NMAT  64
#define STR   68                  // padded LDS row stride; 68%64=4 -> conflict-free panel reads
#define MSZ   (NMAT * STR)        // 4352 floats per LDS matrix buffer
#define NPART 32                  // J-reduction partial buffers

typedef float v2f __attribute__((ext_vector_type(2)));
typedef float v8f __attribute__((ext_vector_type(8)));

// workspace layout (float offsets; 64x64 blocks kept 16KB-aligned for float4 I/O)
#define WS_MEAN  0                // Karcher mean iterate
#define WS_C12   4096             // mean^{1/2}
#define WS_C12I  8192             // mean^{-1/2}
#define WS_W     12288            // sqrt(bias) @ mean^{-1/2}
#define WS_SI    16384            // temp stash for mean^{-1/2} in finalize
#define WS_NU    20480            // step size nu
#define WS_TAU   20481            // tau
#define WS_JPART 24576            // NPART x 4096 partial tangent-mean accumulators

// ---------------- 64x64 GEMM (C = A * B^T) on WMMA f32 16x16x4 --------------
// wave w -> M-row mt = w>>1, N-tiles nt = 2*(w&1)+{0,1}; A-fragment shared.
// Caller must ensure A/B are visible (all call sites follow a synced op);
// trailing barrier makes C visible.
__device__ __forceinline__ void wg_gemm64t(const float* A, const float* B, float* C)
{
    const int tid  = (int)threadIdx.x;
    const int wave = tid >> 5;
    const int lane = tid & 31;
    const int hl   = lane >> 4;   // half-wave
    const int l16  = lane & 15;
#if __has_builtin(__builtin_amdgcn_wmma_f32_16x16x4_f32)
    const int mt    = wave >> 1;
    const int bcol0 = ((wave & 1) * 2) * 16 + l16;
    const int arow  = mt * 16 + l16;
    v8f acc0 = {};
    v8f acc1 = {};
#pragma unroll
    for (int kb = 0; kb < 16; ++kb) {           // K = 64 in steps of 4
        const int k0 = kb * 4 + 2 * hl;         // ISA 16x4 f32 layout: vgpr v, half h -> K=2h+v
        v2f a, b0, b1;
        a.x  = A[arow * STR + k0 + 0];          // contiguous -> ds_load_b64
        a.y  = A[arow * STR + k0 + 1];
        b0.x = B[bcol0 * STR + k0 + 0];         // B^T fragment: row-major read, contiguous
        b0.y = B[bcol0 * STR + k0 + 1];
        b1.x = B[(bcol0 + 16) * STR + k0 + 0];
        b1.y = B[(bcol0 + 16) * STR + k0 + 1];
        acc0 = __builtin_amdgcn_wmma_f32_16x16x4_f32(false, a, false, b0,
                                                     (short)0, acc0, false, false);
        acc1 = __builtin_amdgcn_wmma_f32_16x16x4_f32(false, a, false, b1,
                                                     (short)0, acc1, false, false);
    }
#pragma unroll
    for (int r = 0; r < 8; ++r) {               // C/D layout: row = 16*mt + 8*half + r
        const int crow = (mt * 16 + hl * 8 + r) * STR;
        C[crow + bcol0]      = acc0[r];
        C[crow + bcol0 + 16] = acc1[r];
    }
#else
    for (int p = 0; p < 16; ++p) {              // scalar fallback (compile-safety only)
        const int idx = tid + p * TPB;
        const int i = idx >> 6, j = idx & 63;
        float s = 0.f;
        for (int k = 0; k < NMAT; ++k) s += A[i * STR + k] * B[j * STR + k];
        C[i * STR + j] = s;
    }
#endif
    __syncthreads();
}

// ---------------- helpers ---------------------------------------------------
__device__ __forceinline__ void ld_mat(const float* g, float* s) // global 64x64 -> padded LDS
{
    for (int p = 0; p < 4; ++p) {
        const int q = (int)threadIdx.x + p * TPB;   // float4 index 0..1023
        const int i = q >> 4;
        const int j = (q & 15) << 2;
        const float4 v = ((const float4*)g)[q];
        *(float4*)&s[i * STR + j] = v;
    }
    __syncthreads();
}

__device__ __forceinline__ void st_mat(float* g, const float* s) // padded LDS -> global
{
    for (int p = 0; p < 4; ++p) {
        const int q = (int)threadIdx.x + p * TPB;
        const int i = q >> 4;
        const int j = (q & 15) << 2;
        ((float4*)g)[q] = *(const float4*)&s[i * STR + j];
    }
}

__device__ __forceinline__ float wg_frob2(const float* A, float* red)
{
    float s = 0.f;
    for (int p = 0; p < 16; ++p) {
        const int idx = (int)threadIdx.x + p * TPB;
        const float v = A[(idx >> 6) * STR + (idx & 63)];
        s += v * v;
    }
    red[threadIdx.x] = s;
    __syncthreads();
    for (int off = TPB / 2; off > 0; off >>= 1) {
        if ((int)threadIdx.x < off) red[threadIdx.x] += red[threadIdx.x + off];
        __syncthreads();
    }
    const float r = red[0];
    __syncthreads();
    return r;
}

__device__ __forceinline__ void wg_copy(float* dst, const float* src)
{
    for (int p = 0; p < 4; ++p) {
        const int q = (int)threadIdx.x + p * TPB;
        const int o = (q >> 4) * STR + ((q & 15) << 2);
        *(float4*)&dst[o] = *(const float4*)&src[o];
    }
    __syncthreads();
}

// Coupled Newton-Schulz: A normalized (frob=1, eigs in (0,1]) -> Y=A^{1/2}, Z=A^{-1/2}.
// All operands symmetric, so A*B^T == A*B throughout.
__device__ __forceinline__ void wg_ns_sqrt(const float* A, float* Y, float* Z,
                                           float* T, float* Yn, float* Zn, int iters)
{
    for (int p = 0; p < 16; ++p) {
        const int idx = (int)threadIdx.x + p * TPB;
        const int i = idx >> 6, j = idx & 63, o = i * STR + j;
        Y[o] = A[o];
        Z[o] = (i == j) ? 1.f : 0.f;
    }
    __syncthreads();
    for (int it = 0; it < iters; ++it) {
        wg_gemm64t(Z, Y, T);                         // T = Z*Y
        for (int p = 0; p < 16; ++p) {               // T = 0.5*(3I - T)
            const int idx = (int)threadIdx.x + p * TPB;
            const int i = idx >> 6, j = idx & 63, o = i * STR + j;
            T[o] = 0.5f * (((i == j) ? 3.f : 0.f) - T[o]);
        }
        __syncthreads();
        wg_gemm64t(Y, T, Yn);                        // Y <- Y*T
        wg_gemm64t(T, Z, Zn);                        // Z <- T*Z
        wg_copy(Y, Yn);
        wg_copy(Z, Zn);
    }
}

// ---------------- kernels ---------------------------------------------------

__global__ void k_init(float* ws)
{
    const int t = (int)threadIdx.x;
    for (int idx = t; idx < 4096; idx += TPB) ws[WS_MEAN + idx] = 0.f;
    if (t == 0) { ws[WS_NU] = 1.0f; ws[WS_TAU] = 3.402823466e38f; }
}

__global__ void k_mean_acc(const float* __restrict__ X, float* ws, int batch)
{
    float acc[16];
#pragma unroll
    for (int p = 0; p < 16; ++p) acc[p] = 0.f;
    for (int b = (int)blockIdx.x; b < batch; b += (int)gridDim.x) {
        const int bn = b + (int)gridDim.x;
        if (bn < batch)                                    // gfx1250 global_prefetch_b8
            __builtin_prefetch(X + (size_t)bn * 4096 + threadIdx.x, 0, 1);
        const float* xb = X + (size_t)b * 4096;
#pragma unroll
        for (int p = 0; p < 16; ++p) acc[p] += xb[(int)threadIdx.x + p * TPB];
    }
    const float w = 1.0f / (float)batch;
#pragma unroll
    for (int p = 0; p < 16; ++p)
        atomicAdd(&ws[WS_MEAN + (int)threadIdx.x + p * TPB], acc[p] * w);
}

// Per Karcher iteration: C12 = mean^{1/2}, C12i = mean^{-1/2}; zero J partials.
__global__ void k_prep_iter(float* ws)
{
    __shared__ float sm[6][MSZ];
    __shared__ float red[TPB];
    for (int idx = (int)threadIdx.x; idx < NPART * 4096; idx += TPB)
        ws[WS_JPART + idx] = 0.f;
    ld_mat(ws + WS_MEAN, sm[0]);
    const float s = sqrtf(wg_frob2(sm[0], red));
    const float inv = 1.f / s;
    for (int p = 0; p < 16; ++p) {
        const int o = (((int)threadIdx.x + p * TPB) >> 6) * STR + (((int)threadIdx.x + p * TPB) & 63);
        sm[0][o] *= inv;
    }
    __syncthreads();
    wg_ns_sqrt(sm[0], sm[1], sm[2], sm[3], sm[4], sm[5], 14);
    const float sq = sqrtf(s), isq = rsqrtf(s);
    for (int p = 0; p < 16; ++p) {
        const int idx = (int)threadIdx.x + p * TPB;
        const int o = (idx >> 6) * STR + (idx & 63);
        ws[WS_C12  + idx] = sq  * sm[1][o];
        ws[WS_C12I + idx] = isq * sm[2][o];
    }
}

// Per batch element: M = C12i*X*C12i; logM via 4 NS sqrt levels + Mercator series;
// accumulate into one of NPART partial J buffers.
__global__ void k_log_acc(const float* __restrict__ X, float* ws)
{
    __shared__ float sm[6][MSZ];
    __shared__ float red[TPB];
    ld_mat(X + (size_t)blockIdx.x * 4096, sm[0]);       // X_b (symmetric)
    ld_mat(ws + WS_C12I, sm[1]);                        // C12i (symmetric)
    wg_gemm64t(sm[1], sm[0], sm[2]);                    // T1 = C12i * X
    wg_gemm64t(sm[2], sm[1], sm[3]);                    // M  = T1 * C12i
    const float s = sqrtf(wg_frob2(sm[3], red));
    const float inv = 1.f / s, logs = logf(s);
    for (int p = 0; p < 16; ++p) {
        const int o = (((int)threadIdx.x + p * TPB) >> 6) * STR + (((int)threadIdx.x + p * TPB) & 63);
        sm[3][o] *= inv;                                // A = M/s, eigs in (0,1]
    }
    __syncthreads();
    for (int lev = 0; lev < 4; ++lev) {                 // A <- A^{1/16}
        wg_ns_sqrt(sm[3], sm[4], sm[5], sm[2], sm[0], sm[1], 6);
        wg_copy(sm[3], sm[4]);
    }
    for (int p = 0; p < 16; ++p) {                      // E = A - I
        const int idx = (int)threadIdx.x + p * TPB;
        const int i = idx >> 6, j = idx & 63, o = i * STR + j;
        sm[4][o] = sm[3][o] - ((i == j) ? 1.f : 0.f);
    }
    __syncthreads();
    wg_gemm64t(sm[4], sm[4], sm[2]);                    // E^2
    wg_gemm64t(sm[2], sm[4], sm[0]);                    // E^3
    wg_gemm64t(sm[2], sm[2], sm[1]);                    // E^4
    float* jpart = ws + WS_JPART + (size_t)(blockIdx.x & (NPART - 1)) * 4096;
    for (int p = 0; p < 16; ++p) {
        const int idx = (int)threadIdx.x + p * TPB;
        const int i = idx >> 6, j = idx & 63, o = i * STR + j;
        const float L = sm[4][o] - 0.5f * sm[2][o] + (1.f / 3.f) * sm[0][o] - 0.25f * sm[1][o];
        const float v = 16.f * L + ((i == j) ? logs : 0.f);   // log M = 2^4 log(A^{1/16}) + log(s) I
        atomicAdd(&jpart[idx], v);
    }
}

// J = sum(partials)/B; expJ = expm(nu*J) by scaling-and-squaring;
// mean = C12*expJ*C12; nu/tau step-size update.
__global__ void k_update(float* ws, int batch)
{
    __shared__ float sm[6][MSZ];
    __shared__ float red[TPB];
    __shared__ float snu;
    if (threadIdx.x == 0) snu = ws[WS_NU];
    const float w = 1.f / (float)batch;
    for (int p = 0; p < 16; ++p) {                      // fold NPART partials -> J
        const int idx = (int)threadIdx.x + p * TPB;
        float acc = 0.f;
        for (int q = 0; q < NPART; ++q) acc += ws[WS_JPART + q * 4096 + idx];
        sm[0][(idx >> 6) * STR + (idx & 63)] = acc * w;
    }
    __syncthreads();
    const float crit = sqrtf(wg_frob2(sm[0], red));     // ||J||_F
    const float nu = snu;
    for (int p = 0; p < 16; ++p) {                      // E = nu*J / 2^6
        const int o = (((int)threadIdx.x + p * TPB) >> 6) * STR + (((int)threadIdx.x + p * TPB) & 63);
        sm[1][o] = sm[0][o] * (nu / 64.f);
    }
    __syncthreads();
    wg_gemm64t(sm[1], sm[1], sm[2]);                    // E^2
    wg_gemm64t(sm[2], sm[1], sm[3]);                    // E^3
    wg_gemm64t(sm[2], sm[2], sm[4]);                    // E^4
    for (int p = 0; p < 16; ++p) {                      // T = I+E+E^2/2+E^3/6+E^4/24
        const int idx = (int)threadIdx.x + p * TPB;
        const int i = idx >> 6, j = idx & 63, o = i * STR + j;
        sm[0][o] = ((i == j) ? 1.f : 0.f) + sm[1][o] + 0.5f * sm[2][o]
                 + (1.f / 6.f) * sm[3][o] + (1.f / 24.f) * sm[4][o];
    }
    __syncthreads();
    for (int sq = 0; sq < 6; ++sq) {                    // square back up
        wg_gemm64t(sm[0], sm[0], sm[5]);
        wg_copy(sm[0], sm[5]);
    }
    ld_mat(ws + WS_C12, sm[1]);
    wg_gemm64t(sm[1], sm[0], sm[2]);                    // C12 * expJ
    wg_gemm64t(sm[2], sm[1], sm[3]);                    // * C12
    for (int p = 0; p < 16; ++p) {
        const int idx = (int)threadIdx.x + p * TPB;
        ws[WS_MEAN + idx] = sm[3][(idx >> 6) * STR + (idx & 63)];
    }
    if (threadIdx.x == 0) {
        const float tau = ws[WS_TAU];
        const float h = nu * crit;
        if (h < tau) { ws[WS_NU] = 0.95f * nu; ws[WS_TAU] = h; }
        else         { ws[WS_NU] = 0.5f * nu; }
    }
}

// W = sqrt(bias) * mean^{-1/2}; store W (no transpose needed: apply uses A*B^T).
__global__ void k_finalize(const float* __restrict__ bias, float* ws)
{
    __shared__ float sm[6][MSZ];
    __shared__ float red[TPB];
    // Si = mean^{-1/2}
    ld_mat(ws + WS_MEAN, sm[0]);
    float s = sqrtf(wg_frob2(sm[0], red));
    float inv = 1.f / s;
    for (int p = 0; p < 16; ++p) {
        const int o = (((int)threadIdx.x + p * TPB) >> 6) * STR + (((int)threadIdx.x + p * TPB) & 63);
        sm[0][o] *= inv;
    }
    __syncthreads();
    wg_ns_sqrt(sm[0], sm[1], sm[2], sm[3], sm[4], sm[5], 14);
    {
        const float isq = rsqrtf(s);
        for (int p = 0; p < 16; ++p) {
            const int idx = (int)threadIdx.x + p * TPB;
            ws[WS_SI + idx] = isq * sm[2][(idx >> 6) * STR + (idx & 63)]; // stash Si
        }
    }
    __syncthreads();
    // Sb = bias^{1/2}
    ld_mat(bias, sm[0]);
    s = sqrtf(wg_frob2(sm[0], red));
    inv = 1.f / s;
    for (int p = 0; p < 16; ++p) {
        const int o = (((int)threadIdx.x + p * TPB) >> 6) * STR + (((int)threadIdx.x + p * TPB) & 63);
        sm[0][o] *= inv;
    }
    __syncthreads();
    wg_ns_sqrt(sm[0], sm[1], sm[2], sm[3], sm[4], sm[5], 14);
    {
        const float sq = sqrtf(s);
        for (int p = 0; p < 16; ++p) {
            const int o = (((int)threadIdx.x + p * TPB) >> 6) * STR + (((int)threadIdx.x + p * TPB) & 63);
            sm[1][o] *= sq;                               // Sb in sm[1]
        }
    }
    __syncthreads();
    ld_mat(ws + WS_SI, sm[0]);                            // reload Si (padded)
    wg_gemm64t(sm[1], sm[0], sm[2]);                      // W = Sb * Si (Si symmetric)
    for (int p = 0; p < 16; ++p) {
        const int idx = (int)threadIdx.x + p * TPB;
        ws[WS_W + idx] = sm[2][(idx >> 6) * STR + (idx & 63)];
    }
}

// Y_b = W * X_b * W^T  ==  gemm_t(gemm_t(W, X), W)
__global__ void k_apply(const float* __restrict__ X, const float* __restrict__ ws,
                        float* __restrict__ out)
{
    __shared__ float sm[4][MSZ];
    ld_mat(X + (size_t)blockIdx.x * 4096, sm[0]);
    ld_mat(ws + WS_W, sm[1]);
    wg_gemm64t(sm[1], sm[0], sm[2]);                      // W * X   (X symmetric)
    wg_gemm64t(sm[2], sm[1], sm[3]);                      // * W^T   (true transpose)
    st_mat(out + (size_t)blockIdx.x * 4096, sm[3]);
}

// ---------------- launcher ---------------------------------------------------
extern "C" void kernel_launch(void* const* d_in, const int* in_sizes, int n_in,
                              void* d_out, int out_size, void* d_ws, size_t ws_size,
                              hipStream_t stream)
{
    const float* X    = (const float*)d_in[0];
    const float* bias = (const float*)d_in[1];
    float* out = (float*)d_out;
    float* ws  = (float*)d_ws;
    const int batch = in_sizes[0] / (NMAT * NMAT);   // 8192

    k_init<<<1, TPB, 0, stream>>>(ws);
    k_mean_acc<<<128, TPB, 0, stream>>>(X, ws, batch);
    for (int it = 0; it < 5; ++it) {
        k_prep_iter<<<1, TPB, 0, stream>>>(ws);
        k_log_acc<<<batch, TPB, 0, stream>>>(X, ws);
        k_update<<<1, TPB, 0, stream>>>(ws, batch);
    }
    k_finalize<<<1, TPB, 0, stream>>>(bias, ws);
    k_apply<<<batch, TPB, 0, stream>>>(X, ws, out);
    (void)n_in; (void)out_size; (void)ws_size;
}